// SpeciesAwareRadialBasis_83880711290993
// MI455X (gfx1250) — compile-verified
//
#include <hip/hip_runtime.h>
#include <hip/hip_bf16.h>
#include <math.h>

typedef __attribute__((ext_vector_type(2))) float v2f;
typedef __attribute__((ext_vector_type(4))) float v4f;
typedef __attribute__((ext_vector_type(8))) float v8f;

#define PI_F 3.14159265358979323846f
#define CUTOFF_F 5.0f
#define INV_CUTOFF 0.2f
#define N_ELEM 83

// ---------------------------------------------------------------------------
// Kernel 1: combined[z][k] = sum_c emb[z][c] * dense_w[c][k] + dense_b[k]
// 83x64x8 GEMM done once with V_WMMA_F32_16X16X4_F32.
// 6 waves (192 threads), wave w owns M-tile rows [16w, 16w+16).
// A (16x4 f32) per-lane layout (ISA 7.12.2): lanes 0-15 hold K = k+0,k+1;
// lanes 16-31 hold K = k+2,k+3, M = lane&15.  B (4x16) mirrored: N = lane&15.
// D (16x16 f32): VGPR j = row j (lanes 0-15) / row j+8 (lanes 16-31).
// ---------------------------------------------------------------------------
__global__ __launch_bounds__(192) void precompute_combined(
    const float* __restrict__ emb,   // (83, 64)
    const float* __restrict__ dw,    // (64, 8)
    const float* __restrict__ db,    // (8,)
    float* __restrict__ combined)    // (83, 8) in workspace
{
    const int wave = threadIdx.x >> 5;      // 0..5 -> M tile
    const int lane = threadIdx.x & 31;
    const int half = lane >> 4;             // 0: K+0/1, 1: K+2/3
    const int lidx = lane & 15;
    const int mrow = wave * 16 + lidx;      // A row this lane feeds
    const int mclamp = (mrow < N_ELEM) ? mrow : (N_ELEM - 1);
    const int ncol = lidx;                  // B column (0..15, valid < 8)
    const int nclamp = ncol & 7;

    v8f acc = {0.f, 0.f, 0.f, 0.f, 0.f, 0.f, 0.f, 0.f};
#pragma unroll
    for (int k = 0; k < 64; k += 4) {
        const int ka = k + half * 2;
        float a0 = emb[mclamp * 64 + ka];
        float a1 = emb[mclamp * 64 + ka + 1];
        float b0 = dw[ka * 8 + nclamp];
        float b1 = dw[(ka + 1) * 8 + nclamp];
        v2f a, b;
        a.x = (mrow < N_ELEM) ? a0 : 0.f;
        a.y = (mrow < N_ELEM) ? a1 : 0.f;
        b.x = (ncol < 8) ? b0 : 0.f;
        b.y = (ncol < 8) ? b1 : 0.f;
        // (neg_a, A, neg_b, B, c_mod, C, reuse_a, reuse_b)
        acc = __builtin_amdgcn_wmma_f32_16x16x4_f32(
            false, a, false, b, (short)0, acc, false, false);
    }

    const float bias = db[nclamp];
#pragma unroll
    for (int j = 0; j < 8; ++j) {
        const int M = wave * 16 + j + half * 8;
        if (M < N_ELEM && ncol < 8) combined[M * 8 + ncol] = acc[j] + bias;
    }
}

// ---------------------------------------------------------------------------
// Kernel 2: streaming per-edge kernel.  One wave per edge.
// lane -> lm = lane>>1, k0 = 4*(lane&1); flat out offset = n*128 + lane*4,
// so each lane emits one aligned non-temporal float4 (512 B / wave, coalesced).
// ---------------------------------------------------------------------------
__global__ __launch_bounds__(256) void edge_kernel(
    const float* __restrict__ disp,      // (N, 3)
    const int* __restrict__ zj,          // (N,)
    const float* __restrict__ combined,  // (83, 8)
    const float* __restrict__ tw,        // (4, 8)
    float* __restrict__ out,             // (N, 16, 8)
    int nEdges)
{
    const int wid  = blockIdx.x * (blockDim.x >> 5) + (threadIdx.x >> 5);
    const int lane = threadIdx.x & 31;
    if (wid >= nEdges) return;

    const float dx = disp[3 * wid + 0];
    const float dy = disp[3 * wid + 1];
    const float dz = disp[3 * wid + 2];
    const float r  = sqrtf(dx * dx + dy * dy + dz * dz);
    const float rs = (r > 0.f) ? r : 1.f;
    const float inv = 1.f / rs;
    const float x = dx * inv, y = dy * inv, z = dz * inv;
    const float x2 = x * x, y2 = y * y, z2 = z * z;

    const float s3   = 1.7320508075688772f;   // sqrt(3)
    const float s15  = 3.8729833462074170f;   // sqrt(15)
    const float s5_8 = 0.7905694150420949f;   // sqrt(5/8)
    const float s3_8 = 0.6123724356957945f;   // sqrt(3/8)

    float Y[16];
    Y[0]  = 1.f;
    Y[1]  = x;
    Y[2]  = y;
    Y[3]  = z;
    Y[4]  = s3 * x * y;
    Y[5]  = s3 * y * z;
    Y[6]  = 0.5f * (3.f * z2 - 1.f);
    Y[7]  = s3 * x * z;
    Y[8]  = 0.5f * s3 * (x2 - y2);
    Y[9]  = s5_8 * y * (3.f * x2 - y2);
    Y[10] = s15 * x * y * z;
    Y[11] = s3_8 * y * (5.f * z2 - 1.f);
    Y[12] = 0.5f * z * (5.f * z2 - 3.f);
    Y[13] = s3_8 * x * (5.f * z2 - 1.f);
    Y[14] = 0.5f * s15 * z * (x2 - y2);
    Y[15] = s5_8 * x * (x2 - 3.f * y2);

    const int lm = lane >> 1;
    const int k0 = (lane & 1) * 4;

    // branchless select (cndmask chain) keeps the wave convergent
    float Ysel = Y[0];
#pragma unroll
    for (int i = 1; i < 16; ++i) Ysel = (lm == i) ? Y[i] : Ysel;

    const int l = (lm == 0) ? 0 : (lm < 4) ? 1 : (lm < 9) ? 2 : 3;

    const float cut = (r < CUTOFF_F)
                        ? 0.5f * (cosf(PI_F * r * INV_CUTOFF) + 1.f)
                        : 0.f;

    const int zi = zj[wid];
    const v4f t4 = *(const v4f*)(combined + zi * 8 + k0);  // 16B aligned
    const v4f w0 = *(const v4f*)(tw + k0);                 // tensor_w[0]
    const v4f wl = *(const v4f*)(tw + l * 8 + k0);         // tensor_w[DEG[lm]]

    const float tt[4]  = {t4.x, t4.y, t4.z, t4.w};
    const float tg0[4] = {w0.x, w0.y, w0.z, w0.w};
    const float twl[4] = {wl.x, wl.y, wl.z, wl.w};

    float o[4];
#pragma unroll
    for (int j = 0; j < 4; ++j) {
        const float kk  = (float)(k0 + j + 1);
        const float pa  = PI_F * (r * kk * INV_CUTOFF);
        const float snc = (r > 0.f) ? (sinf(pa) / pa) : 1.f;  // jnp.sinc
        const float rb  = snc * cut;
        const float trb = tt[j] * rb;
        const float g   = 1.f / (1.f + expf(-(trb * tg0[j]))); // Y[0]==1
        o[j] = trb * twl[j] * Ysel * g;
    }

    v4f ov;
    ov.x = o[0]; ov.y = o[1]; ov.z = o[2]; ov.w = o[3];
    __builtin_nontemporal_store(ov, (v4f*)out + (size_t)wid * 32 + lane);
}

extern "C" void kernel_launch(void* const* d_in, const int* in_sizes, int n_in,
                              void* d_out, int out_size, void* d_ws, size_t ws_size,
                              hipStream_t stream) {
    const float* disp = (const float*)d_in[0];  // (N,3) f32
    const int*   zj   = (const int*)d_in[1];    // (N,) i32
    const float* emb  = (const float*)d_in[2];  // (83,64) f32
    const float* dw   = (const float*)d_in[3];  // (64,8) f32
    const float* db   = (const float*)d_in[4];  // (8,) f32
    const float* tw   = (const float*)d_in[5];  // (4,8) f32
    float* out        = (float*)d_out;          // (N,16,8) f32
    float* combined   = (float*)d_ws;           // needs 83*8*4 = 2656 B

    const int N = in_sizes[0] / 3;

    precompute_combined<<<dim3(1), dim3(192), 0, stream>>>(emb, dw, db, combined);

    const int wavesPerBlock = 8;                 // 256 threads = 8 wave32
    const int blocks = (N + wavesPerBlock - 1) / wavesPerBlock;
    edge_kernel<<<dim3(blocks), dim3(256), 0, stream>>>(disp, zj, combined, tw,
                                                        out, N);
}